// RetinaNetLoss_86526411145412
// MI455X (gfx1250) — compile-verified
//
#include <hip/hip_runtime.h>
#include <hip/hip_bf16.h>

typedef float v2f __attribute__((ext_vector_type(2)));
typedef float v8f __attribute__((ext_vector_type(8)));

namespace {
constexpr int NG   = 50;        // GT boxes per image
constexpr int NBI  = 8;         // batch
constexpr int KTOT = 120087;    // total anchors
constexpr int BK   = NBI * KTOT;          // 960,696
constexpr int KBLK = (KTOT + 255) / 256;  // 470 blocks per image
constexpr int NBB  = KBLK * NBI;          // 3,760 bbox partials
constexpr int ETOT = 8 * BK;              // 7,685,568 cls elements
constexpr int NCB  = (ETOT + 255) / 256;  // 30,022 cls partials
constexpr int NBB_PAD = ((NBB + 63) / 64) * 64;  // 3,776
constexpr int NCB_PAD = ((NCB + 63) / 64) * 64;  // 30,080
}

__constant__ int c_OFF[6]  = {0, 90000, 112500, 118125, 119646, 120087};
__constant__ int c_FW[5]   = {100, 50, 25, 13, 7};
__constant__ int c_STR[5]  = {8, 16, 32, 64, 128};
__constant__ int c_EOFF[6] = {0, 5760000, 7200000, 7560000, 7657344, 7685568};
__constant__ int c_KL[5]   = {90000, 22500, 5625, 1521, 441};

// ---- helpers ---------------------------------------------------------------

__device__ inline void decode_k(int k, int& lvl, int& a, int& yy, int& xx) {
  int l = 0;
  if (k >= c_OFF[1]) l = 1;
  if (k >= c_OFF[2]) l = 2;
  if (k >= c_OFF[3]) l = 3;
  if (k >= c_OFF[4]) l = 4;
  lvl = l;
  int kin = k - c_OFF[l];
  int loc = kin / 9;
  a = kin - loc * 9;
  int fw = c_FW[l];
  yy = loc / fw;
  xx = loc - yy * fw;
}

// RetinaNet anchor, f64 base math like numpy, cast to f32
__device__ inline void anchor_box(int lvl, int a, int yy, int xx,
                                  float& x0, float& y0, float& x1, float& y1) {
#pragma clang fp contract(off)
  const double HR[3] = {0.7071067811865476, 1.0, 1.4142135623730951}; // sqrt(ratio)
  const double SC[3] = {4.0, 5.039684199579493, 6.3496042078727974};  // 4*2^(j/3)
  int ri = a / 3, si = a - ri * 3;
  double st = (double)c_STR[lvl];
  double hr = HR[ri];
  double wr = 1.0 / hr;
  double wsz = wr * SC[si] * st;
  double hsz = hr * SC[si] * st;
  double sx = (double)(xx * c_STR[lvl]);
  double sy = (double)(yy * c_STR[lvl]);
  x0 = (float)(sx - wsz * 0.5);
  y0 = (float)(sy - hsz * 0.5);
  x1 = (float)(sx + wsz * 0.5);
  y1 = (float)(sy + hsz * 0.5);
}

// contract(off): pass1 and pass2 must produce bit-identical IoUs (equality test)
__device__ inline float iou_one(float ax0, float ay0, float ax1, float ay1,
                                float bx0, float by0, float bx1, float by1) {
#pragma clang fp contract(off)
  float area_a = (ax1 - ax0) * (ay1 - ay0);
  float area_b = (bx1 - bx0) * (by1 - by0);
  float ltx = fmaxf(ax0, bx0), lty = fmaxf(ay0, by0);
  float rbx = fminf(ax1, bx1), rby = fminf(ay1, by1);
  float w = fmaxf(rbx - ltx, 0.f), h = fmaxf(rby - lty, 0.f);
  float inter = w * h;
  return inter / (area_a + area_b - inter);
}

// wave32 shuffle reduction (deterministic, fixed order)
__device__ __forceinline__ float wave_sum_f(float v) {
  v += __shfl_xor(v, 16, 32);
  v += __shfl_xor(v, 8, 32);
  v += __shfl_xor(v, 4, 32);
  v += __shfl_xor(v, 2, 32);
  v += __shfl_xor(v, 1, 32);
  return v;
}
__device__ __forceinline__ unsigned wave_sum_u(unsigned v) {
  v += __shfl_xor(v, 16, 32);
  v += __shfl_xor(v, 8, 32);
  v += __shfl_xor(v, 4, 32);
  v += __shfl_xor(v, 2, 32);
  v += __shfl_xor(v, 1, 32);
  return v;
}

// ---- pass 1: per-GT max IoU over all inside anchors ------------------------

__global__ void k_gtmax(const float* __restrict__ tb, unsigned* __restrict__ gtb) {
  __shared__ float    s_gt[NG * 4];
  __shared__ unsigned s_max[NG];
  int b = blockIdx.y;
  int tid = threadIdx.x;
  for (int i = tid; i < NG * 4; i += 256) s_gt[i] = tb[b * NG * 4 + i];
  if (tid < NG) s_max[tid] = 0u;
  __syncthreads();
  int k = blockIdx.x * 256 + tid;
  if (k < KTOT) {
    int lvl, a, yy, xx;
    decode_k(k, lvl, a, yy, xx);
    float x0, y0, x1, y1;
    anchor_box(lvl, a, yy, xx, x0, y0, x1, y1);
    bool inside = (x0 >= 0.f) && (y0 >= 0.f) && (x1 <= 800.f) && (y1 <= 800.f);
    if (inside) {
      for (int g = 0; g < NG; ++g) {
        float v = iou_one(x0, y0, x1, y1,
                          s_gt[g * 4 + 0], s_gt[g * 4 + 1],
                          s_gt[g * 4 + 2], s_gt[g * 4 + 3]);
        if (v > 0.f) atomicMax(&s_max[g], __float_as_uint(v));  // >0 floats order like uints
      }
    }
  }
  __syncthreads();
  if (tid < NG) {
    unsigned m = s_max[tid];
    if (m) atomicMax(&gtb[b * NG + tid], m);
  }
}

// ---- pass 2: matching + targets + fused bbox L1 loss -----------------------

__global__ void k_match(const float* __restrict__ tb, const int* __restrict__ tc,
                        const unsigned* __restrict__ gtb,
                        const float* __restrict__ r0, const float* __restrict__ r1,
                        const float* __restrict__ r2, const float* __restrict__ r3,
                        const float* __restrict__ r4,
                        unsigned char* __restrict__ meta,
                        float* __restrict__ pbb, unsigned* __restrict__ cnt) {
  __shared__ float    s_gt[NG * 4];
  __shared__ float    s_gm[NG];
  __shared__ int      s_cl[NG];
  __shared__ float    s_wf[8];
  __shared__ unsigned s_wu[8];
  int b = blockIdx.y;
  int tid = threadIdx.x;
  for (int i = tid; i < NG * 4; i += 256) s_gt[i] = tb[b * NG * 4 + i];
  if (tid < NG) {
    s_gm[tid] = __uint_as_float(gtb[b * NG + tid]);
    s_cl[tid] = tc[b * NG + tid];
  }
  __syncthreads();
  int k = blockIdx.x * 256 + tid;
  float lb = 0.f;
  unsigned c1 = 0;
  if (k < KTOT) {
    int lvl, a, yy, xx;
    decode_k(k, lvl, a, yy, xx);
    float ax0, ay0, ax1, ay1;
    anchor_box(lvl, a, yy, xx, ax0, ay0, ax1, ay1);
    bool inside = (ax0 >= 0.f) && (ay0 >= 0.f) && (ax1 <= 800.f) && (ay1 <= 800.f);
    float best = -2.f;
    int am = 0, lqg = -1;
    for (int g = 0; g < NG; ++g) {
      float v = iou_one(ax0, ay0, ax1, ay1,
                        s_gt[g * 4 + 0], s_gt[g * 4 + 1],
                        s_gt[g * 4 + 2], s_gt[g * 4 + 3]);
      v = inside ? v : -1.f;                 // mask outside anchors
      if (v > best) { best = v; am = g; }    // first-occurrence argmax
      float gm = s_gm[g];
      if (gm > 0.f && v == gm) lqg = g;      // last gt wins low-quality ties
    }
    bool anylq = (lqg >= 0);
    int matched = anylq ? lqg : am;
    int label = anylq ? 1 : (best >= 0.5f ? 1 : (best >= 0.4f ? -1 : 0));
    bool pos = (label == 1) && inside;
    bool neg = (label == 0) && inside;
    bool lwf = pos || neg;
    meta[b * KTOT + k] =
        (unsigned char)((pos ? (s_cl[matched] & 15) : 0) | (lwf ? 16 : 0));
    c1 = lwf ? 1u : 0u;
    if (pos) {
#pragma clang fp contract(off)
      float aw = ax1 - ax0, ah = ay1 - ay0;
      float acx = ax0 + 0.5f * aw, acy = ay0 + 0.5f * ah;
      float gx0 = s_gt[matched * 4 + 0], gy0 = s_gt[matched * 4 + 1];
      float gx1 = s_gt[matched * 4 + 2], gy1 = s_gt[matched * 4 + 3];
      float gw = gx1 - gx0, gh = gy1 - gy0;
      float gcx = gx0 + 0.5f * gw, gcy = gy0 + 0.5f * gh;
      float d0 = (gcx - acx) / aw, d1 = (gcy - acy) / ah;
      float d2 = logf(gw / aw), d3 = logf(gh / ah);
      const float* rp = (lvl == 0) ? r0 : (lvl == 1) ? r1 : (lvl == 2) ? r2
                        : (lvl == 3) ? r3 : r4;
      int H = c_FW[lvl];
      int plane = H * H;
      int base = ((b * 36 + a * 4) * H + yy) * H + xx;
      lb = fabsf(rp[base] - d0) + fabsf(rp[base + plane] - d1) +
           fabsf(rp[base + 2 * plane] - d2) + fabsf(rp[base + 3 * plane] - d3);
    }
  }
  // wave32 shuffle reduce, then 8 wave-leaders via tiny LDS step
  float wf = wave_sum_f(lb);
  unsigned wu = wave_sum_u(c1);
  int wave = tid >> 5, lane = tid & 31;
  if (lane == 0) { s_wf[wave] = wf; s_wu[wave] = wu; }
  __syncthreads();
  if (tid == 0) {
    float rs = 0.f;
    unsigned ru = 0;
    for (int i = 0; i < 8; ++i) { rs += s_wf[i]; ru += s_wu[i]; }
    pbb[blockIdx.y * gridDim.x + blockIdx.x] = rs;
    atomicAdd(cnt, ru);
  }
}

// ---- pass 3: classification BCE with tile(lw,8) weight indexing ------------

__global__ void k_cls(const float* __restrict__ c0, const float* __restrict__ c1,
                      const float* __restrict__ c2, const float* __restrict__ c3,
                      const float* __restrict__ c4,
                      const unsigned char* __restrict__ meta,
                      float* __restrict__ pcls) {
  __shared__ float s_wf[8];
  int j = blockIdx.x * 256 + threadIdx.x;
  float acc = 0.f;
  if (j < ETOT) {
    int lvl = 0;
    if (j >= c_EOFF[1]) lvl = 1;
    if (j >= c_EOFF[2]) lvl = 2;
    if (j >= c_EOFF[3]) lvl = 3;
    if (j >= c_EOFF[4]) lvl = 4;
    int e = j - c_EOFF[lvl];
    int KL = c_KL[lvl];
    int N = NBI * KL;
    int n = e >> 3, c = e & 7;
    int widx = e % N;                 // (8n+c) % N, the tile(lw,8) quirk
    int wb = widx / KL, wk = widx - wb * KL;
    float w = (meta[wb * KTOT + c_OFF[lvl] + wk] & 16) ? 1.f : 0.f;
    if (w != 0.f) {
      int bb = n / KL, kin = n - bb * KL;
      int L = meta[bb * KTOT + c_OFF[lvl] + kin] & 15;
      float t = (c == L) ? 1.f : 0.f;
      int loc = kin / 9, a = kin - loc * 9;
      int H = c_FW[lvl];
      int yy = loc / H, xx = loc - yy * H;
      const float* cp = (lvl == 0) ? c0 : (lvl == 1) ? c1 : (lvl == 2) ? c2
                        : (lvl == 3) ? c3 : c4;
      float x = cp[((bb * 72 + a * 8 + c) * H + yy) * H + xx];
      acc = (fmaxf(x, 0.f) - x * t + log1pf(expf(-fabsf(x)))) * w;
    }
  }
  float wf = wave_sum_f(acc);
  int wave = threadIdx.x >> 5, lane = threadIdx.x & 31;
  if (lane == 0) s_wf[wave] = wf;
  __syncthreads();
  if (threadIdx.x == 0) {
    float rs = 0.f;
    for (int i = 0; i < 8; ++i) rs += s_wf[i];
    pcls[blockIdx.x] = rs;
  }
}

// ---- final: deterministic reduction of partials via V_WMMA_F32_16X16X4_F32 -
// Arrays are zero-padded to multiples of 64, so the loop is branch-free:
// each lane does one aligned b64 load (2 floats of the 16x4 f32 A-tile),
// A x ones(4x16) accumulates exact-f32 row sums into the C tile.

__device__ __forceinline__ float wmma_sum(const float* __restrict__ p, int n64, int lane) {
  v2f ones; ones[0] = 1.f; ones[1] = 1.f;
  v8f acc = {0.f, 0.f, 0.f, 0.f, 0.f, 0.f, 0.f, 0.f};
  const float* q = p + (lane << 1);
  for (int i = 0; i < n64; i += 64) {
    v2f a = *(const v2f*)(q + i);
#if __has_builtin(__builtin_amdgcn_wmma_f32_16x16x4_f32)
    acc = __builtin_amdgcn_wmma_f32_16x16x4_f32(
        /*neg_a=*/false, a, /*neg_b=*/false, ones,
        /*c_mod=*/(short)0, acc, /*reuse_a=*/false, /*reuse_b=*/false);
#else
    acc[0] += a[0] + a[1];  // fallback sums only lanes l and l^16, not all 32
#endif
  }
  float s = acc[0] + acc[1] + acc[2] + acc[3] + acc[4] + acc[5] + acc[6] + acc[7];
  s += __shfl_xor(s, 16, 32);  // lanes 0-15 hold M=0..7 sums, 16-31 hold M=8..15
  return s;
}

__global__ void k_final(const float* __restrict__ pc, const float* __restrict__ pb,
                        const unsigned* __restrict__ cnt, float* __restrict__ out) {
  int lane = threadIdx.x;
  float sc = wmma_sum(pc, NCB_PAD, lane);
  float sb = wmma_sum(pb, NBB_PAD, lane);
  if (lane == 0) {
    float ns = (float)(*cnt);
    out[0] = sc / ns;
    out[1] = sb / ns;
  }
}

// ---- launch ----------------------------------------------------------------

extern "C" void kernel_launch(void* const* d_in, const int* in_sizes, int n_in,
                              void* d_out, int out_size, void* d_ws, size_t ws_size,
                              hipStream_t stream) {
  // dict order: cls0,reg0,cls1,reg1,...,cls4,reg4,target_boxes,target_class_ids
  const float* cls[5];
  const float* reg[5];
  for (int i = 0; i < 5; ++i) {
    cls[i] = (const float*)d_in[2 * i];
    reg[i] = (const float*)d_in[2 * i + 1];
  }
  const float* tb = (const float*)d_in[10];
  const int*   tc = (const int*)d_in[11];

  auto al256 = [](size_t x) { return (x + 255) & ~(size_t)255; };
  char* w = (char*)d_ws;
  size_t o = 0;
  unsigned char* meta = (unsigned char*)(w + o); o += al256((size_t)BK);
  float* pcls = (float*)(w + o);                 o += al256((size_t)NCB_PAD * 4);
  float* pbb  = (float*)(w + o);                 o += al256((size_t)NBB_PAD * 4);
  unsigned* gtb = (unsigned*)(w + o);            o += al256((size_t)NBI * NG * 4);
  unsigned* cnt = (unsigned*)(w + o);            o += 256;

  hipMemsetAsync(gtb, 0, (size_t)NBI * NG * 4, stream);
  hipMemsetAsync(cnt, 0, 4, stream);
  // zero the padding tails so the WMMA reduction can run branch-free
  hipMemsetAsync(pcls + NCB, 0, (size_t)(NCB_PAD - NCB) * 4, stream);
  hipMemsetAsync(pbb + NBB, 0, (size_t)(NBB_PAD - NBB) * 4, stream);

  dim3 grid2(KBLK, NBI);
  k_gtmax<<<grid2, 256, 0, stream>>>(tb, gtb);
  k_match<<<grid2, 256, 0, stream>>>(tb, tc, gtb,
                                     reg[0], reg[1], reg[2], reg[3], reg[4],
                                     meta, pbb, cnt);
  k_cls<<<NCB, 256, 0, stream>>>(cls[0], cls[1], cls[2], cls[3], cls[4],
                                 meta, pcls);
  k_final<<<1, 32, 0, stream>>>(pcls, pbb, cnt, (float*)d_out);
}